// VectorQuantizer_86887188398519
// MI455X (gfx1250) — compile-verified
//
#include <hip/hip_runtime.h>
#include <math.h>

#define DDIM 64
#define KDIM 512
#define NROWS (64 * 32 * 32)   // 65536 rows
#define WAVES_PER_BLOCK 8
#define THREADS (WAVES_PER_BLOCK * 32)

typedef float v2f __attribute__((ext_vector_type(2)));
typedef float v8f __attribute__((ext_vector_type(8)));

// ---------------------------------------------------------------- init ws ---
__global__ void vq_init_kernel(double* sq_sum, unsigned int* counts) {
    int t = threadIdx.x;
    if (t == 0) *sq_sum = 0.0;
    if (t < KDIM) counts[t] = 0u;
}

// ------------------------------------------------------------------- main ---
__launch_bounds__(THREADS)
__global__ void vq_main_kernel(const float* __restrict__ x,
                               const float* __restrict__ emb,
                               float* __restrict__ quant_out,
                               float* __restrict__ idx_out,
                               double* __restrict__ sq_sum,
                               unsigned int* __restrict__ counts) {
    __shared__ float lds_emb[KDIM * DDIM];   // 128 KB, fits 320 KB WGP LDS
    __shared__ float lds_e2[KDIM];           // 2 KB

    const int tid  = threadIdx.x;
    const int lane = tid & 31;
    const int wave = tid >> 5;
    const int lh   = lane & 15;   // position within 16-lane half
    const int hi   = lane >> 4;   // 0 = lanes 0-15, 1 = lanes 16-31

    // Cooperative load of full codebook into LDS (float4, coalesced).
    {
        const float4* src = (const float4*)emb;
        float4*       dst = (float4*)lds_emb;
        for (int i = tid; i < (KDIM * DDIM) / 4; i += THREADS) dst[i] = src[i];
    }
    __syncthreads();

    // ||e_k||^2 per code.
    for (int k = tid; k < KDIM; k += THREADS) {
        const float* e = &lds_emb[k * DDIM];
        float s = 0.f;
#pragma unroll
        for (int d = 0; d < DDIM; ++d) s += e[d] * e[d];
        lds_e2[k] = s;
    }
    __syncthreads();

    const int row_base = (blockIdx.x * WAVES_PER_BLOCK + wave) * 16;

    // A fragments for the 16-row tile: V_WMMA_F32_16X16X4_F32 A layout (16x4):
    // lanes 0-15 hold K=0,1 ; lanes 16-31 hold K=2,3  -> float2 at col 4s+2*hi.
    v2f a[16];
    {
        const v2f* xr2 = (const v2f*)(x + (size_t)(row_base + lh) * DDIM + 2 * hi);
#pragma unroll
        for (int s = 0; s < 16; ++s) a[s] = xr2[2 * s];
    }

    float bestv[8];
    int   besti[8];
#pragma unroll
    for (int r = 0; r < 8; ++r) { bestv[r] = INFINITY; besti[r] = 0; }

    // Sweep the 32 column tiles of 16 codes each.
    for (int j = 0; j < 32; ++j) {
        const int col = j * 16;
        v8f c = {0.f, 0.f, 0.f, 0.f, 0.f, 0.f, 0.f, 0.f};

        // B fragment (4x16) from LDS: lane holds code col+lh, K pair 2*hi.
        const v2f* er2 = (const v2f*)(lds_emb + (size_t)(col + lh) * DDIM + 2 * hi);
#pragma unroll
        for (int s = 0; s < 16; ++s) {
            v2f b = er2[2 * s];
            c = __builtin_amdgcn_wmma_f32_16x16x4_f32(
                    /*neg_a=*/false, a[s], /*neg_b=*/false, b,
                    /*c_mod=*/(short)0, c, /*reuse_a=*/false, /*reuse_b=*/false);
        }

        // distance surrogate: ||e||^2 - 2 * (x . e)   (||x||^2 is row-constant)
        const float e2c  = lds_e2[col + lh];
        const int   kidx = col + lh;
#pragma unroll
        for (int r = 0; r < 8; ++r) {
            float v = e2c - 2.0f * c[r];
            if (v < bestv[r]) { bestv[r] = v; besti[r] = kidx; }
        }
    }

    // Butterfly argmin across the 16 lanes of each half (ties -> lower index,
    // matching jnp.argmin first-hit semantics).
#pragma unroll
    for (int m = 1; m < 16; m <<= 1) {
#pragma unroll
        for (int r = 0; r < 8; ++r) {
            float ov = __shfl_xor(bestv[r], m, 32);
            int   oi = __shfl_xor(besti[r], m, 32);
            if (ov < bestv[r] || (ov == bestv[r] && oi < besti[r])) {
                bestv[r] = ov; besti[r] = oi;
            }
        }
    }

    // Lanes 0 / 16 own rows 0-7 / 8-15: write indices + histogram.
    if (lh == 0) {
#pragma unroll
        for (int r = 0; r < 8; ++r) {
            int row = row_base + hi * 8 + r;
            idx_out[row] = (float)besti[r];
            atomicAdd(&counts[besti[r]], 1u);
        }
    }

    // Gather quantized rows from LDS, write out, accumulate sum((q-x)^2).
    float lsq = 0.f;
#pragma unroll
    for (int m = 0; m < 16; ++m) {
        int k = __shfl(besti[m & 7], (m < 8) ? 0 : 16, 32);
        const v2f* q2 = (const v2f*)(lds_emb + (size_t)k * DDIM);
        const v2f* x2 = (const v2f*)(x + (size_t)(row_base + m) * DDIM);
        v2f q  = q2[lane];
        v2f xx = x2[lane];
        ((v2f*)(quant_out + (size_t)(row_base + m) * DDIM))[lane] = q;
        float d0 = q.x - xx.x, d1 = q.y - xx.y;
        lsq += d0 * d0 + d1 * d1;
    }
#pragma unroll
    for (int m = 1; m < 32; m <<= 1) lsq += __shfl_xor(lsq, m, 32);
    if (lane == 0) atomicAdd(sq_sum, (double)lsq);   // global_atomic_add_f64
}

// --------------------------------------------------------------- finalize ---
__global__ void vq_final_kernel(const double* __restrict__ sq_sum,
                                const unsigned int* __restrict__ counts,
                                float* __restrict__ loss_out,
                                float* __restrict__ perp_out) {
    __shared__ float red[KDIM];
    int t = threadIdx.x;
    float p = (float)counts[t] * (1.0f / (float)NROWS);
    red[t] = p * logf(p + 1e-10f);
    __syncthreads();
    for (int s = KDIM / 2; s > 0; s >>= 1) {
        if (t < s) red[t] += red[t + s];
        __syncthreads();
    }
    if (t == 0) {
        // loss = emb_latent + 0.25*inp_latent; both equal mean((q-x)^2) fwd.
        *loss_out = 1.25f * (float)(*sq_sum / (double)((size_t)NROWS * DDIM));
        *perp_out = expf(-red[0]);
    }
}

// ----------------------------------------------------------------- launch ---
extern "C" void kernel_launch(void* const* d_in, const int* in_sizes, int n_in,
                              void* d_out, int out_size, void* d_ws, size_t ws_size,
                              hipStream_t stream) {
    (void)in_sizes; (void)n_in; (void)out_size; (void)ws_size;

    const float* x   = (const float*)d_in[0];   // [64,32,32,64] f32
    const float* emb = (const float*)d_in[1];   // [512,64] f32

    float* out      = (float*)d_out;
    float* quant    = out;                            // N*D
    float* loss     = out + (size_t)NROWS * DDIM;     // 1
    float* perp     = loss + 1;                       // 1
    float* idx_grid = perp + 1;                       // N (as float)

    double*       sq     = (double*)d_ws;
    unsigned int* counts = (unsigned int*)((char*)d_ws + 64);

    vq_init_kernel<<<1, KDIM, 0, stream>>>(sq, counts);
    vq_main_kernel<<<NROWS / (16 * WAVES_PER_BLOCK), THREADS, 0, stream>>>(
        x, emb, quant, idx_grid, sq, counts);
    vq_final_kernel<<<1, KDIM, 0, stream>>>(sq, counts, loss, perp);
}